// GAT_17145509446185
// MI455X (gfx1250) — compile-verified
//
#include <hip/hip_runtime.h>
#include <hip/hip_bf16.h>
#include <math.h>

typedef __attribute__((ext_vector_type(2))) float v2f;
typedef __attribute__((ext_vector_type(8))) float v8f;

#define NPG   90          // nodes per graph
#define INF_D 90          // input feature dim
#define HDIM  64          // hidden dim
#define MP    96          // padded node count (6 tiles of 16)
#define HS    66          // hbuf row stride (floats)
#define PS    97          // Pbuf row stride (floats)
#define EPG   8010        // edges per graph (90*89)
#define ZROW  11520       // 90*128 floats per graph row of Z

__device__ __forceinline__ v8f wmma4(v2f a, v2f b, v8f c) {
#if defined(__HIP_DEVICE_COMPILE__)
  // (neg_a, A, neg_b, B, c_mod, C, reuse_a, reuse_b) -> v_wmma_f32_16x16x4_f32
  return __builtin_amdgcn_wmma_f32_16x16x4_f32(false, a, false, b, (short)0, c,
                                               false, false);
#else
  return c;
#endif
}

// ---------------------------------------------------------------------------
// K0: mean of edge_attr (single workgroup reduction)
// ---------------------------------------------------------------------------
__global__ __launch_bounds__(256) void edge_mean_kernel(
    const float* __restrict__ ea, int E, float* __restrict__ meanp) {
  __shared__ float red[256];
  float s = 0.f;
  for (int i = threadIdx.x; i < E; i += 256) s += ea[i];
  red[threadIdx.x] = s;
  __syncthreads();
  for (int off = 128; off > 0; off >>= 1) {
    if (threadIdx.x < off) red[threadIdx.x] += red[threadIdx.x + off];
    __syncthreads();
  }
  if (threadIdx.x == 0) meanp[0] = red[0] / (float)E;
}

// ---------------------------------------------------------------------------
// K1: full 2-layer GAT for one graph per workgroup. Writes Z[g][m*128+c].
// ---------------------------------------------------------------------------
__global__ __launch_bounds__(256) void gat_graph_kernel(
    const float* __restrict__ x,    // [N, 90]
    const float* __restrict__ ea,   // [E, 1]
    const float* __restrict__ W1, const float* __restrict__ as1,
    const float* __restrict__ ad1, const float* __restrict__ We1,
    const float* __restrict__ ae1, const float* __restrict__ b1,
    const float* __restrict__ W2, const float* __restrict__ as2,
    const float* __restrict__ ad2, const float* __restrict__ We2,
    const float* __restrict__ ae2, const float* __restrict__ b2,
    const float* __restrict__ meanp,
    float* __restrict__ Z)          // [B, 11520]
{
  const int g    = blockIdx.x;
  const int tid  = threadIdx.x;
  const int w    = tid >> 5;        // wave id (wave32, 8 waves)
  const int lane = tid & 31;
  const int lm   = lane & 15;       // M/N index within 16
  const int lk   = (lane >> 4) * 2; // K sub-offset {0,2}

  __shared__ float hbuf[MP * HS];   // W1 -> h1 -> x1 -> h2 (in place)
  __shared__ float Pbuf[MP * PS];   // x -> P1 -> W2 -> P2
  __shared__ float asv[MP], adv[MP];
  __shared__ float scal[1];

  const float* xg  = x  + (size_t)g * NPG * INF_D;
  const float* eag = ea + (size_t)g * EPG;
  const float  mean = *meanp;
  float*       Zg  = Z + (size_t)g * ZROW;

  v8f acc[3];

  // ================= Layer 1 =================
  // stage x_g (zero-padded) into Pbuf [96][97] and W1 into hbuf [96][66]
  for (int idx = tid; idx < MP * PS; idx += 256) {
    const int r = idx / PS, c = idx % PS;
    Pbuf[idx] = (r < NPG && c < INF_D) ? xg[r * INF_D + c] : 0.f;
  }
  for (int idx = tid; idx < MP * HS; idx += 256) {
    const int r = idx / HS, c = idx % HS;
    hbuf[idx] = (r < INF_D && c < HDIM) ? W1[r * HDIM + c] : 0.f;
  }
  __syncthreads();

  // h1 = pad(x_g) @ W1  (96x96 * 96x64), operands entirely in LDS
  #pragma unroll
  for (int q = 0; q < 3; ++q) {
    const int t  = w + 8 * q;       // tile 0..23
    const int mt = t >> 2, nt = t & 3;
    const int m = mt * 16 + lm;     // row 0..95
    const int n = nt * 16 + lm;     // col 0..63
    v8f c = {};
    #pragma unroll 4
    for (int kk = 0; kk < 24; ++kk) {
      const int kb = kk * 4 + lk;   // 0..94 (even)
      v2f a, b;
      a.x = Pbuf[m * PS + kb];
      a.y = Pbuf[m * PS + kb + 1];
      b.x = hbuf[kb * HS + n];
      b.y = hbuf[(kb + 1) * HS + n];
      c = wmma4(a, b, c);
    }
    acc[q] = c;
  }
  __syncthreads();
  #pragma unroll
  for (int q = 0; q < 3; ++q) {
    const int t  = w + 8 * q;
    const int mt = t >> 2, nt = t & 3;
    const int n = nt * 16 + lm;
    const int row0 = mt * 16 + 8 * (lane >> 4);
    #pragma unroll
    for (int r = 0; r < 8; ++r) hbuf[(row0 + r) * HS + n] = acc[q][r];
  }
  __syncthreads();

  // attention scalars
  if (tid < MP) {
    float s = 0.f, d = 0.f;
    for (int k = 0; k < HDIM; ++k) {
      const float hv = hbuf[tid * HS + k];
      s += hv * as1[k];
      d += hv * ad1[k];
    }
    asv[tid] = s; adv[tid] = d;
  }
  if (tid == 0) {
    float c = 0.f;
    for (int k = 0; k < HDIM; ++k) c += We1[k] * ae1[k];
    scal[0] = c;
  }
  __syncthreads();
  float ce = scal[0];

  // P[i][j] = leaky_relu(a_s[j] + a_d[i] + ce*ea(j->i))
  for (int idx = tid; idx < MP * MP; idx += 256) {
    const int i = idx / MP, j = idx % MP;   // i = dst, j = src
    if (i < NPG && j < NPG) {
      const float e = (i == j) ? ce * mean
                               : ce * eag[j * 89 + ((i > j) ? i - 1 : i)];
      const float s = asv[j] + adv[i] + e;
      Pbuf[i * PS + j] = (s > 0.f) ? s : 0.2f * s;
    }
  }
  __syncthreads();

  // row softmax over j (one thread per row); zero the padding
  if (tid < MP) {
    const int i = tid;
    if (i < NPG) {
      float mx = -1e30f;
      for (int j = 0; j < NPG; ++j) mx = fmaxf(mx, Pbuf[i * PS + j]);
      float sum = 0.f;
      for (int j = 0; j < NPG; ++j) {
        const float e = expf(Pbuf[i * PS + j] - mx);
        Pbuf[i * PS + j] = e;
        sum += e;
      }
      const float inv = 1.f / sum;
      for (int j = 0; j < NPG; ++j) Pbuf[i * PS + j] *= inv;
      for (int j = NPG; j < PS; ++j) Pbuf[i * PS + j] = 0.f;
    } else {
      for (int j = 0; j < PS; ++j) Pbuf[i * PS + j] = 0.f;
    }
  }
  __syncthreads();

  // out1 = P @ h1 ; x1 = relu(out1 + b1), written in place into hbuf and Z
  #pragma unroll
  for (int q = 0; q < 3; ++q) {
    const int t  = w + 8 * q;
    const int mt = t >> 2, nt = t & 3;
    const int m = mt * 16 + lm, n = nt * 16 + lm;
    v8f c = {};
    #pragma unroll 4
    for (int kk = 0; kk < 24; ++kk) {
      const int kb = kk * 4 + lk;
      v2f a, b;
      a.x = Pbuf[m * PS + kb];
      a.y = Pbuf[m * PS + kb + 1];
      b.x = hbuf[kb * HS + n];
      b.y = hbuf[(kb + 1) * HS + n];
      c = wmma4(a, b, c);
    }
    acc[q] = c;
  }
  __syncthreads();
  #pragma unroll
  for (int q = 0; q < 3; ++q) {
    const int t  = w + 8 * q;
    const int mt = t >> 2, nt = t & 3;
    const int n = nt * 16 + lm;
    const int row0 = mt * 16 + 8 * (lane >> 4);
    #pragma unroll
    for (int r = 0; r < 8; ++r) {
      const int m = row0 + r;
      float v = 0.f;
      if (m < NPG) {
        v = fmaxf(acc[q][r] + b1[n], 0.f);
        Zg[m * 128 + n] = v;                  // x1 half of z
      }
      hbuf[m * HS + n] = v;                   // x1 (pad rows zeroed)
    }
  }

  // ================= Layer 2 =================
  // stage W2 into Pbuf region as [64][66]
  __syncthreads();
  for (int idx = tid; idx < HDIM * HS; idx += 256) {
    const int r = idx / HS, c = idx % HS;
    Pbuf[idx] = (c < HDIM) ? W2[r * HDIM + c] : 0.f;
  }
  __syncthreads();

  // h2 = x1 @ W2   (96x64 * 64x64), operands in LDS
  #pragma unroll
  for (int q = 0; q < 3; ++q) {
    const int t  = w + 8 * q;
    const int mt = t >> 2, nt = t & 3;
    const int m = mt * 16 + lm, n = nt * 16 + lm;
    v8f c = {};
    #pragma unroll 4
    for (int kk = 0; kk < 16; ++kk) {
      const int kb = kk * 4 + lk;             // 0..62
      v2f a, b;
      a.x = hbuf[m * HS + kb];
      a.y = hbuf[m * HS + kb + 1];
      b.x = Pbuf[kb * HS + n];
      b.y = Pbuf[(kb + 1) * HS + n];
      c = wmma4(a, b, c);
    }
    acc[q] = c;
  }
  __syncthreads();
  #pragma unroll
  for (int q = 0; q < 3; ++q) {
    const int t  = w + 8 * q;
    const int mt = t >> 2, nt = t & 3;
    const int n = nt * 16 + lm;
    const int row0 = mt * 16 + 8 * (lane >> 4);
    #pragma unroll
    for (int r = 0; r < 8; ++r) hbuf[(row0 + r) * HS + n] = acc[q][r];
  }
  __syncthreads();

  if (tid < MP) {
    float s = 0.f, d = 0.f;
    for (int k = 0; k < HDIM; ++k) {
      const float hv = hbuf[tid * HS + k];
      s += hv * as2[k];
      d += hv * ad2[k];
    }
    asv[tid] = s; adv[tid] = d;
  }
  if (tid == 0) {
    float c = 0.f;
    for (int k = 0; k < HDIM; ++k) c += We2[k] * ae2[k];
    scal[0] = c;
  }
  __syncthreads();
  ce = scal[0];

  for (int idx = tid; idx < MP * MP; idx += 256) {
    const int i = idx / MP, j = idx % MP;
    if (i < NPG && j < NPG) {
      const float e = (i == j) ? ce * mean
                               : ce * eag[j * 89 + ((i > j) ? i - 1 : i)];
      const float s = asv[j] + adv[i] + e;
      Pbuf[i * PS + j] = (s > 0.f) ? s : 0.2f * s;
    }
  }
  __syncthreads();

  if (tid < MP) {
    const int i = tid;
    if (i < NPG) {
      float mx = -1e30f;
      for (int j = 0; j < NPG; ++j) mx = fmaxf(mx, Pbuf[i * PS + j]);
      float sum = 0.f;
      for (int j = 0; j < NPG; ++j) {
        const float e = expf(Pbuf[i * PS + j] - mx);
        Pbuf[i * PS + j] = e;
        sum += e;
      }
      const float inv = 1.f / sum;
      for (int j = 0; j < NPG; ++j) Pbuf[i * PS + j] *= inv;
      for (int j = NPG; j < PS; ++j) Pbuf[i * PS + j] = 0.f;
    } else {
      for (int j = 0; j < PS; ++j) Pbuf[i * PS + j] = 0.f;
    }
  }
  __syncthreads();

  // out2 = P2 @ h2 ; x2 = relu(out2 + b2) -> Z (cols 64..127)
  #pragma unroll
  for (int q = 0; q < 3; ++q) {
    const int t  = w + 8 * q;
    const int mt = t >> 2, nt = t & 3;
    const int m = mt * 16 + lm, n = nt * 16 + lm;
    v8f c = {};
    #pragma unroll 4
    for (int kk = 0; kk < 24; ++kk) {
      const int kb = kk * 4 + lk;
      v2f a, b;
      a.x = Pbuf[m * PS + kb];
      a.y = Pbuf[m * PS + kb + 1];
      b.x = hbuf[kb * HS + n];
      b.y = hbuf[(kb + 1) * HS + n];
      c = wmma4(a, b, c);
    }
    const int row0 = mt * 16 + 8 * (lane >> 4);
    #pragma unroll
    for (int r = 0; r < 8; ++r) {
      const int m2 = row0 + r;
      if (m2 < NPG) Zg[m2 * 128 + 64 + n] = fmaxf(c[r] + b2[n], 0.f);
    }
  }
}

// ---------------------------------------------------------------------------
// K2: MLP head. Z[256x11520] @ Wl1[11520x64] + bl1 -> relu -> @Wl2[64x2]+bl2
//     -> log_softmax. One workgroup per 16 output rows; K staged through LDS
//     in 64-wide chunks with coalesced float4 loads.
// ---------------------------------------------------------------------------
__global__ __launch_bounds__(256) void mlp_head_kernel(
    const float* __restrict__ Z,
    const float* __restrict__ Wl1, const float* __restrict__ bl1,
    const float* __restrict__ Wl2, const float* __restrict__ bl2,
    float* __restrict__ out) {
  const int mt   = blockIdx.x;      // 0..15
  const int tid  = threadIdx.x;
  const int w    = tid >> 5;
  const int lane = tid & 31;
  const int lm   = lane & 15;
  const int lk   = (lane >> 4) * 2;

  __shared__ float ldsA[16 * HS];      // Z tile:   16 x 64, stride 66
  __shared__ float ldsB[64 * HS];      // Wl1 tile: 64 x 64, stride 66
  __shared__ float red[8 * 32 * 32];   // 32 KB cross-wave partials
  __shared__ float hdd[16 * 64];
  __shared__ float lg[16 * 2];

  v8f acc[4];
  #pragma unroll
  for (int nt = 0; nt < 4; ++nt) { v8f z = {}; acc[nt] = z; }

  // A-stage indices: thread -> one float4 of the 16x64 tile
  const int ar = (tid * 4) >> 6;       // 0..15
  const int ac = (tid * 4) & 63;       // 0,4,..,60

  for (int kc = 0; kc < ZROW / 64; ++kc) {   // 180 chunks
    const int k0 = kc * 64;
    // stage A (16x64): one float4 per thread, coalesced
    {
      const float4 va =
          *(const float4*)&Z[(size_t)(mt * 16 + ar) * ZROW + k0 + ac];
      ldsA[ar * HS + ac + 0] = va.x;
      ldsA[ar * HS + ac + 1] = va.y;
      ldsA[ar * HS + ac + 2] = va.z;
      ldsA[ar * HS + ac + 3] = va.w;
    }
    // stage B (64x64): four float4 per thread, coalesced
    #pragma unroll
    for (int u = 0; u < 4; ++u) {
      const int t4 = (tid + u * 256) * 4;    // 0..4092
      const int br = t4 >> 6, bc = t4 & 63;
      const float4 vb = *(const float4*)&Wl1[(size_t)(k0 + br) * 64 + bc];
      ldsB[br * HS + bc + 0] = vb.x;
      ldsB[br * HS + bc + 1] = vb.y;
      ldsB[br * HS + bc + 2] = vb.z;
      ldsB[br * HS + bc + 3] = vb.w;
    }
    __syncthreads();
    // wave w consumes k-steps w and w+8 of the 16 in this chunk
    #pragma unroll
    for (int h = 0; h < 2; ++h) {
      const int kb = (w + 8 * h) * 4 + lk;   // 0..62
      v2f a;
      a.x = ldsA[lm * HS + kb];
      a.y = ldsA[lm * HS + kb + 1];
      #pragma unroll
      for (int nt = 0; nt < 4; ++nt) {
        const int n = nt * 16 + lm;
        v2f b;
        b.x = ldsB[kb * HS + n];
        b.y = ldsB[(kb + 1) * HS + n];
        acc[nt] = wmma4(a, b, acc[nt]);
      }
    }
    __syncthreads();
  }

  #pragma unroll
  for (int nt = 0; nt < 4; ++nt)
    #pragma unroll
    for (int r = 0; r < 8; ++r)
      red[(w * 32 + nt * 8 + r) * 32 + lane] = acc[nt][r];
  __syncthreads();

  #pragma unroll
  for (int rep = 0; rep < 4; ++rep) {
    const int idx = tid + rep * 256;   // 0..1023
    const int s = idx >> 5, l = idx & 31;
    float sum = 0.f;
    #pragma unroll
    for (int w2 = 0; w2 < 8; ++w2) sum += red[(w2 * 32 + s) * 32 + l];
    const int nt = s >> 3, r = s & 7;
    const int row = r + 8 * (l >> 4);
    const int col = nt * 16 + (l & 15);
    hdd[row * 64 + col] = fmaxf(sum + bl1[col], 0.f);
  }
  __syncthreads();

  if (tid < 32) {
    const int row = tid >> 1, cls = tid & 1;
    float s = bl2[cls];
    for (int k = 0; k < 64; ++k) s += hdd[row * 64 + k] * Wl2[k * 2 + cls];
    lg[row * 2 + cls] = s;
  }
  __syncthreads();

  if (tid < 16) {
    const float l0 = lg[tid * 2], l1 = lg[tid * 2 + 1];
    const float mx = fmaxf(l0, l1);
    const float lse = mx + logf(expf(l0 - mx) + expf(l1 - mx));
    const int gr = mt * 16 + tid;
    out[gr * 2]     = l0 - lse;
    out[gr * 2 + 1] = l1 - lse;
  }
}

// ---------------------------------------------------------------------------
extern "C" void kernel_launch(void* const* d_in, const int* in_sizes, int n_in,
                              void* d_out, int out_size, void* d_ws,
                              size_t ws_size, hipStream_t stream) {
  const float* x   = (const float*)d_in[0];
  const float* ea  = (const float*)d_in[2];
  const float* W1  = (const float*)d_in[4];
  const float* as1 = (const float*)d_in[5];
  const float* ad1 = (const float*)d_in[6];
  const float* We1 = (const float*)d_in[7];
  const float* ae1 = (const float*)d_in[8];
  const float* b1  = (const float*)d_in[9];
  const float* W2  = (const float*)d_in[10];
  const float* as2 = (const float*)d_in[11];
  const float* ad2 = (const float*)d_in[12];
  const float* We2 = (const float*)d_in[13];
  const float* ae2 = (const float*)d_in[14];
  const float* b2  = (const float*)d_in[15];
  const float* Wl1 = (const float*)d_in[16];
  const float* bl1 = (const float*)d_in[17];
  const float* Wl2 = (const float*)d_in[18];
  const float* bl2 = (const float*)d_in[19];

  float* ws    = (float*)d_ws;
  float* meanp = ws;        // ws[0]: mean(edge_attr)
  float* Z     = ws + 64;   // 256-byte aligned; [256][11520] floats

  const int E = in_sizes[2];

  edge_mean_kernel<<<1, 256, 0, stream>>>(ea, E, meanp);
  gat_graph_kernel<<<256, 256, 0, stream>>>(x, ea, W1, as1, ad1, We1, ae1, b1,
                                            W2, as2, ad2, We2, ae2, b2, meanp,
                                            Z);
  mlp_head_kernel<<<16, 256, 0, stream>>>(Z, Wl1, bl1, Wl2, bl2,
                                          (float*)d_out);
}